// SoftmaxRuleLayer_42348377539208
// MI455X (gfx1250) — compile-verified
//
#include <hip/hip_runtime.h>
#include <hip/hip_bf16.h>

#define B_  32
#define D_  1024
#define R_  2048

typedef float v2f __attribute__((ext_vector_type(2)));
typedef float v8f __attribute__((ext_vector_type(8)));

// ---------------------------------------------------------------------------
// Helper: insert (v,i) into running top-2 with jax.lax.top_k tie-breaking
// (larger value wins; on equal value, lower index wins).
// ---------------------------------------------------------------------------
__device__ __forceinline__ void top2_insert(float v, int i,
                                            float& v1, int& i1,
                                            float& v2, int& i2) {
    bool beats1 = (v > v1) || (v == v1 && i < i1);
    if (beats1) {
        v2 = v1; i2 = i1;
        v1 = v;  i1 = i;
    } else {
        bool beats2 = (v > v2) || (v == v2 && i < i2);
        if (beats2) { v2 = v; i2 = i; }
    }
}

// ---------------------------------------------------------------------------
// Kernel 1: per-rule top-2 fact indices + aggregator softmax + sigmoid(strength)
// One wave32 per rule; 8 waves per 256-thread block.
// (softmax -> top_k == top_k of logits; softmax values unused downstream)
// ---------------------------------------------------------------------------
__global__ void rule_prep_kernel(const float* __restrict__ fact_logits,   // [R,D]
                                 const float* __restrict__ agg_logits,    // [R,3]
                                 const float* __restrict__ strength,      // [R]
                                 int*   __restrict__ ruleIdx,             // [R,2]
                                 float* __restrict__ ruleW)               // [R,4] w0,w1,w2,sigma
{
    const int wavesPerBlock = blockDim.x >> 5;
    const int r    = blockIdx.x * wavesPerBlock + (threadIdx.x >> 5);
    const int lane = threadIdx.x & 31;
    if (r >= R_) return;

    float v1 = -__builtin_inff(), v2 = -__builtin_inff();
    int   i1 = 0x7fffffff,        i2 = 0x7fffffff;

    const float* row = fact_logits + (size_t)r * D_;
    #pragma unroll 4
    for (int j = lane; j < D_; j += 32) {
        top2_insert(row[j], j, v1, i1, v2, i2);
    }
    // wave32 xor-tree merge of per-lane top-2 lists
    #pragma unroll
    for (int m = 16; m >= 1; m >>= 1) {
        float ov1 = __shfl_xor(v1, m, 32);
        int   oi1 = __shfl_xor(i1, m, 32);
        float ov2 = __shfl_xor(v2, m, 32);
        int   oi2 = __shfl_xor(i2, m, 32);
        top2_insert(ov1, oi1, v1, i1, v2, i2);
        top2_insert(ov2, oi2, v1, i1, v2, i2);
    }

    if (lane == 0) {
        ruleIdx[2 * r + 0] = i1;
        ruleIdx[2 * r + 1] = i2;
        float a0 = agg_logits[3 * r + 0];
        float a1 = agg_logits[3 * r + 1];
        float a2 = agg_logits[3 * r + 2];
        float mx = fmaxf(a0, fmaxf(a1, a2));
        float e0 = expf(a0 - mx), e1 = expf(a1 - mx), e2 = expf(a2 - mx);
        float inv = 1.0f / (e0 + e1 + e2);
        ruleW[4 * r + 0] = e0 * inv;
        ruleW[4 * r + 1] = e1 * inv;
        ruleW[4 * r + 2] = e2 * inv;
        ruleW[4 * r + 3] = 1.0f / (1.0f + expf(-strength[r]));
    }
}

// ---------------------------------------------------------------------------
// Kernel 2: proj = facts @ W_proj^T via V_WMMA_F32_16X16X4_F32.
// K-sliced for memory-level parallelism: wave w handles N-tile (w & 127) and
// K-slice (w >> 7); partial results go to projPart[slice][B][R] and are summed
// deterministically in gate_ln_kernel. One B-fragment load feeds two WMMAs
// (c0: samples 0-15, c1: samples 16-31), so W_proj streams exactly once.
// ISA f32 16x4 A layout: lane l holds A[l%16][ (l/16)*2 + vgpr ];
// B (4x16) symmetric:    lane l holds B[(l/16)*2 + vgpr][ l%16 ].
// C/D layout: vgpr v, lane l -> row (v + (l/16)*8), col (l%16).
// ---------------------------------------------------------------------------
__global__ void gemm_wmma_kernel(const float* __restrict__ facts,     // [B,D]
                                 const float* __restrict__ Wp,        // [R,D]
                                 float* __restrict__ projPart,        // [nslice,B,R]
                                 int klen)                            // D_/nslice
{
    const int wave  = (blockIdx.x * blockDim.x + threadIdx.x) >> 5;
    const int lane  = threadIdx.x & 31;

    const int ntile = wave & 127;          // 0..127  -> n0 = ntile*16
    const int slice = wave >> 7;           // 0..nslice-1
    const int n0    = ntile * 16;
    const int kbeg  = slice * klen;

    const int rlo   = lane & 15;           // 0..15
    const int khalf = (lane >> 4) * 2;     // 0 or 2

    const float* aptr0 = facts + (size_t)(rlo)      * D_ + kbeg + khalf;  // A rows 0-15
    const float* aptr1 = facts + (size_t)(16 + rlo) * D_ + kbeg + khalf;  // A rows 16-31
    const float* bptr  = Wp    + (size_t)(n0 + rlo) * D_ + kbeg + khalf;  // B col = rule

    v8f c0 = {};
    v8f c1 = {};
    for (int k = 0; k < klen; k += 16) {
        // WGP-scope prefetch: pull the W_proj stream into all cache levels
        __builtin_prefetch(bptr + k + 512, 0, 3);
        #pragma unroll
        for (int kk = 0; kk < 16; kk += 4) {
            v2f b  = *(const v2f*)(bptr  + k + kk);
            v2f a0 = *(const v2f*)(aptr0 + k + kk);
            v2f a1 = *(const v2f*)(aptr1 + k + kk);
            // (neg_a, A, neg_b, B, c_mod, C, reuse_a, reuse_b)
            c0 = __builtin_amdgcn_wmma_f32_16x16x4_f32(false, a0, false, b,
                                                       (short)0, c0, false, false);
            c1 = __builtin_amdgcn_wmma_f32_16x16x4_f32(false, a1, false, b,
                                                       (short)0, c1, false, false);
        }
    }

    float* outp = projPart + (size_t)slice * B_ * R_;
    const int col   = n0 + rlo;
    const int rbase = (lane >> 4) * 8;
    #pragma unroll
    for (int v = 0; v < 8; ++v) {
        outp[(size_t)(rbase + v)      * R_ + col] = c0[v];
        outp[(size_t)(rbase + v + 16) * R_ + col] = c1[v];
    }
}

// ---------------------------------------------------------------------------
// Kernel 3: rule_act[b,r] = mixed_aggregators * sigmoid(strength)
// One thread per (b,r).
// ---------------------------------------------------------------------------
__global__ void rule_act_kernel(const float* __restrict__ facts,    // [B,D]
                                const int*   __restrict__ ruleIdx,  // [R,2]
                                const float* __restrict__ ruleW,    // [R,4]
                                float* __restrict__ ract)           // [B,R]
{
    const int gid = blockIdx.x * blockDim.x + threadIdx.x;  // 0 .. B*R-1
    const int b = gid >> 11;          // / R_
    const int r = gid & (R_ - 1);

    const float f0 = facts[(size_t)b * D_ + ruleIdx[2 * r + 0]];
    const float f1 = facts[(size_t)b * D_ + ruleIdx[2 * r + 1]];
    const float w0 = ruleW[4 * r + 0];
    const float w1 = ruleW[4 * r + 1];
    const float w2 = ruleW[4 * r + 2];
    const float sg = ruleW[4 * r + 3];

    const float and_a = f0 * f1;
    const float or_a  = 1.0f - (1.0f - f0) * (1.0f - f1);
    const float k_a   = (f0 + f1) / (2.0f + 1e-8f);

    ract[gid] = (w0 * and_a + w1 * or_a + w2 * k_a) * sg;
}

// ---------------------------------------------------------------------------
// Kernel 4: sum K-slice partials + per-sample top-8 rule gating + LayerNorm.
// One 256-thread block per sample b; whole row lives in LDS.
// ---------------------------------------------------------------------------
__global__ void gate_ln_kernel(const float* __restrict__ projPart, // [nslice,B,R]
                               int nslice,
                               const float* __restrict__ ract,     // [B,R]
                               const float* __restrict__ gamma,    // [R]
                               const float* __restrict__ beta,     // [R]
                               float* __restrict__ out)            // [B,R]
{
    const int b   = blockIdx.x;
    const int tid = threadIdx.x;          // 256 threads

    __shared__ float s_pre[R_];
    __shared__ float s_act[R_];
    __shared__ unsigned char s_ch[R_];
    __shared__ float s_val[256];
    __shared__ int   s_idx[256];

    // deterministic fixed-order sum of the K-slice partials
    for (int i = tid; i < R_; i += 256) {
        float acc = 0.0f;
        for (int s = 0; s < nslice; ++s) {
            acc += projPart[(size_t)s * B_ * R_ + (size_t)b * R_ + i];
        }
        s_pre[i] = acc;
        s_act[i] = ract[(size_t)b * R_ + i];
        s_ch[i]  = 0;
    }
    __syncthreads();

    // 8 sequential argmax passes (tie -> lower index, matching jax top_k gate)
    for (int t = 0; t < 8; ++t) {
        float bv = -__builtin_inff();
        int   bi = 0x7fffffff;
        for (int i = tid; i < R_; i += 256) {
            if (!s_ch[i]) {
                float v = s_act[i];
                if (v > bv || (v == bv && i < bi)) { bv = v; bi = i; }
            }
        }
        s_val[tid] = bv; s_idx[tid] = bi;
        __syncthreads();
        for (int s = 128; s > 0; s >>= 1) {
            if (tid < s) {
                float ov = s_val[tid + s]; int oi = s_idx[tid + s];
                if (ov > s_val[tid] || (ov == s_val[tid] && oi < s_idx[tid])) {
                    s_val[tid] = ov; s_idx[tid] = oi;
                }
            }
            __syncthreads();
        }
        if (tid == 0 && s_idx[0] < R_) s_ch[s_idx[0]] = 1;
        __syncthreads();
    }

    // pre = proj + rule_act * gate
    for (int i = tid; i < R_; i += 256) {
        if (s_ch[i]) s_pre[i] += s_act[i];
    }
    __syncthreads();

    // mean
    float sum = 0.0f;
    for (int i = tid; i < R_; i += 256) sum += s_pre[i];
    s_val[tid] = sum;
    __syncthreads();
    for (int s = 128; s > 0; s >>= 1) {
        if (tid < s) s_val[tid] += s_val[tid + s];
        __syncthreads();
    }
    const float mu = s_val[0] * (1.0f / R_);
    __syncthreads();

    // variance (two-pass, matches reference)
    float ss = 0.0f;
    for (int i = tid; i < R_; i += 256) { float d = s_pre[i] - mu; ss += d * d; }
    s_val[tid] = ss;
    __syncthreads();
    for (int s = 128; s > 0; s >>= 1) {
        if (tid < s) s_val[tid] += s_val[tid + s];
        __syncthreads();
    }
    const float inv = rsqrtf(s_val[0] * (1.0f / R_) + 1e-5f);

    for (int i = tid; i < R_; i += 256) {
        out[(size_t)b * R_ + i] = (s_pre[i] - mu) * inv * gamma[i] + beta[i];
    }
}

// ---------------------------------------------------------------------------
// Launch
// ---------------------------------------------------------------------------
extern "C" void kernel_launch(void* const* d_in, const int* in_sizes, int n_in,
                              void* d_out, int out_size, void* d_ws, size_t ws_size,
                              hipStream_t stream) {
    const float* facts        = (const float*)d_in[0];   // [B,D]
    const float* fact_logits  = (const float*)d_in[1];   // [R,D]
    const float* agg_logits   = (const float*)d_in[2];   // [R,3]
    const float* strength     = (const float*)d_in[3];   // [R]
    const float* W_proj       = (const float*)d_in[4];   // [R,D]
    const float* ln_gamma     = (const float*)d_in[5];   // [R]
    const float* ln_beta      = (const float*)d_in[6];   // [R]
    float* out = (float*)d_out;                          // [B,R]

    // Choose K-slice count from available workspace (deterministic per run):
    //   need = 48KB rule params + 256KB ract + nslice*256KB proj partials
    const size_t kFixed = 49152 + (size_t)B_ * R_ * 4;
    int nslice = 1;
    if (ws_size >= kFixed + 4 * (size_t)B_ * R_ * 4) nslice = 4;
    else if (ws_size >= kFixed + 2 * (size_t)B_ * R_ * 4) nslice = 2;

    // Workspace layout
    char* ws = (char*)d_ws;
    float* ruleW    = (float*)(ws);                               // R*4 f32 (32 KB)
    int*   ruleIdx  = (int*)  (ws + (size_t)R_ * 4 * 4);          // R*2 i32 (16 KB)
    float* ract     = (float*)(ws + 49152);                       // B*R f32 (256 KB)
    float* projPart = (float*)(ws + kFixed);                      // nslice*B*R f32

    // 1) per-rule prep: 2048 rules, 8 waves/block -> 256 blocks
    rule_prep_kernel<<<R_ / 8, 256, 0, stream>>>(fact_logits, agg_logits, strength,
                                                 ruleIdx, ruleW);

    // 2) GEMM: 128*nslice waves -> 16*nslice blocks of 8 waves
    gemm_wmma_kernel<<<16 * nslice, 256, 0, stream>>>(facts, W_proj, projPart,
                                                      D_ / nslice);

    // 3) rule activations: B*R threads
    rule_act_kernel<<<(B_ * R_) / 256, 256, 0, stream>>>(facts, ruleIdx, ruleW, ract);

    // 4) slice-sum + gate + layernorm: one block per sample
    gate_ln_kernel<<<B_, 256, 0, stream>>>(projPart, nslice, ract,
                                           ln_gamma, ln_beta, out);
}